// Reference_1151051235362
// MI455X (gfx1250) — compile-verified
//
#include <hip/hip_runtime.h>
#include <hip/hip_bf16.h>

typedef __attribute__((ext_vector_type(2))) float v2f;
typedef __attribute__((ext_vector_type(8))) float v8f;

// Column mapping within a 576-wide row: feature (u, i) with i in 0..8 the
// flattened irrep index (l=0:m=0 | l=1:m=0..2 | l=2:m=0..4) lives at
// col = kSegBase[i] + u*kSegDim[i] + kSegM[i]   (MUL=64, OFF={0,1,4,9})
__device__ __constant__ int kSegBase[9] = {0, 64, 64, 64, 256, 256, 256, 256, 256};
__device__ __constant__ int kSegDim[9]  = {1, 3, 3, 3, 5, 5, 5, 5, 5};
__device__ __constant__ int kSegM[9]    = {0, 0, 1, 2, 0, 1, 2, 3, 4};

// ---------------------------------------------------------------------------
// Regenerate the module-level CG constants exactly as numpy RandomState(0)
// would: MT19937 + Marsaglia polar gaussians, consumed in PATHS order, /3.0,
// cast to f32, scattered into a dense [9][9][9] tensor (zeros elsewhere).
// Single-thread init kernel; deterministic & idempotent -> graph-safe.
// ---------------------------------------------------------------------------
__global__ void gen_cg_kernel(float* __restrict__ cg) {
    if (threadIdx.x != 0 || blockIdx.x != 0) return;

    unsigned mt[624];
    mt[0] = 0u; // seed 0
    for (int i = 1; i < 624; ++i)
        mt[i] = 1812433253u * (mt[i - 1] ^ (mt[i - 1] >> 30)) + (unsigned)i;
    int mti = 624;

    auto rk_u32 = [&]() -> unsigned {
        if (mti >= 624) {
            for (int k = 0; k < 624; ++k) {
                unsigned y = (mt[k] & 0x80000000u) | (mt[(k + 1) % 624] & 0x7fffffffu);
                mt[k] = mt[(k + 397) % 624] ^ (y >> 1) ^ ((y & 1u) ? 2567483615u : 0u);
            }
            mti = 0;
        }
        unsigned y = mt[mti++];
        y ^= (y >> 11);
        y ^= (y << 7)  & 2636928640u;
        y ^= (y << 15) & 4022730752u;
        y ^= (y >> 18);
        return y;
    };
    auto rk_double = [&]() -> double {
        unsigned a = rk_u32() >> 5, b = rk_u32() >> 6;
        return (a * 67108864.0 + b) / 9007199254740992.0;
    };
    double gcache = 0.0; bool has_g = false;
    auto next_gauss = [&]() -> double {
        if (has_g) { has_g = false; return gcache; }
        double x1, x2, r2;
        do {
            x1 = 2.0 * rk_double() - 1.0;
            x2 = 2.0 * rk_double() - 1.0;
            r2 = x1 * x1 + x2 * x2;
        } while (r2 >= 1.0 || r2 == 0.0);
        double f = sqrt(-2.0 * log(r2) / r2);
        gcache = f * x1; has_g = true;
        return f * x2;
    };

    for (int t = 0; t < 729; ++t) cg[t] = 0.0f;
    const int OFFv[3] = {0, 1, 4};
    for (int l1 = 0; l1 <= 2; ++l1)
        for (int l2 = 0; l2 <= 2; ++l2)
            for (int l3 = 0; l3 <= 2; ++l3) {
                int lo = (l1 > l2) ? (l1 - l2) : (l2 - l1);
                if (l3 < lo || l3 > l1 + l2) continue;
                for (int a = 0; a < 2 * l1 + 1; ++a)
                    for (int b = 0; b < 2 * l2 + 1; ++b)
                        for (int c = 0; c < 2 * l3 + 1; ++c)
                            cg[((OFFv[l1] + a) * 9 + (OFFv[l2] + b)) * 9 + (OFFv[l3] + c)] =
                                (float)(next_gauss() / 3.0);
            }
}

// ---------------------------------------------------------------------------
// Main kernel: one wave32 per edge.
//   W[i][k] = sum_j sh[j] * CG[i][j][k]            (per-lane VALU, CG in LDS)
//   edge_out[64x9] = Xe[64x9] @ W[9x9]             (12x v_wmma_f32_16x16x4_f32)
//   atomic scatter-add into out[dst]               (global_atomic_add_f32)
// WMMA fragment layouts (wave32, 32-bit, h=lane>>4, n=lane&15):
//   A 16x4 : lane holds M=n, VGPR v holds K = 2h+v        (ISA table)
//   B 4x16 : lane holds N=n, VGPR v holds K = 2h+v        (row-in-half pattern)
//   C 16x16: VGPR r holds (M = r + 8h, N = n)             (ISA table)
// ---------------------------------------------------------------------------
__global__ __launch_bounds__(256) void tp_scatter_kernel(
        const float* __restrict__ x, const float* __restrict__ sh,
        const int* __restrict__ src, const int* __restrict__ dst,
        const float* __restrict__ cg, float* __restrict__ out, int E) {
    __shared__ float sCG[729];
    for (int t = threadIdx.x; t < 729; t += blockDim.x) sCG[t] = cg[t];
    __syncthreads();

    const int wave = threadIdx.x >> 5;
    const int lane = threadIdx.x & 31;
    const int e = blockIdx.x * (blockDim.x >> 5) + wave;
    if (e >= E) return;              // wave-uniform: EXEC stays all-ones below

    const int h = lane >> 4;
    const int n = lane & 15;

    const int snode = src[e];
    const int dnode = dst[e];
    const float* xrow = x + (long long)snode * 576;
    float* orow = out + (long long)dnode * 576;

    // sh broadcast (same address across lanes -> cache broadcast)
    float shv[9];
#pragma unroll
    for (int j = 0; j < 9; ++j) shv[j] = sh[(long long)e * 9 + j];

    // Build B fragments (3 K-steps of 4), rows i>=9 and cols k>=9 are zero pad
    v2f Bf[3];
#pragma unroll
    for (int ks = 0; ks < 3; ++ks) {
#pragma unroll
        for (int v = 0; v < 2; ++v) {
            const int i = 4 * ks + 2 * h + v;
            float acc = 0.0f;
            if (i < 9 && n < 9) {
#pragma unroll
                for (int j = 0; j < 9; ++j)
                    acc += shv[j] * sCG[(i * 9 + j) * 9 + n];
            }
            Bf[ks][v] = acc;
        }
    }

#pragma unroll
    for (int mt = 0; mt < 4; ++mt) {
        const int u = 16 * mt + n;   // A row handled by this lane
        v8f c = {0.f, 0.f, 0.f, 0.f, 0.f, 0.f, 0.f, 0.f};
#pragma unroll
        for (int ks = 0; ks < 3; ++ks) {
            v2f a;
#pragma unroll
            for (int v = 0; v < 2; ++v) {
                const int i = 4 * ks + 2 * h + v;
                a[v] = (i < 9) ? xrow[kSegBase[i] + u * kSegDim[i] + kSegM[i]] : 0.0f;
            }
            c = __builtin_amdgcn_wmma_f32_16x16x4_f32(
                    false, a, false, Bf[ks], (short)0, c, false, false);
        }
        // Scatter-add D fragment: element (M = r + 8h, N = n)
        if (n < 9) {
            const int base = kSegBase[n];
            const int dim  = kSegDim[n];
            const int mm   = kSegM[n];
#pragma unroll
            for (int r = 0; r < 8; ++r) {
                const int uo = 16 * mt + r + 8 * h;
                unsafeAtomicAdd(&orow[base + uo * dim + mm], c[r]);
            }
        }
    }
}

extern "C" void kernel_launch(void* const* d_in, const int* in_sizes, int n_in,
                              void* d_out, int out_size, void* d_ws, size_t ws_size,
                              hipStream_t stream) {
    const float* x   = (const float*)d_in[0];
    const float* sh  = (const float*)d_in[1];
    const int*   src = (const int*)d_in[2];
    const int*   dst = (const int*)d_in[3];
    float* out = (float*)d_out;
    float* cg  = (float*)d_ws;                 // 729 floats of scratch
    const int E = in_sizes[2];

    hipMemsetAsync(d_out, 0, (size_t)out_size * sizeof(float), stream);
    gen_cg_kernel<<<1, 32, 0, stream>>>(cg);

    const int wavesPerBlock = 8;               // 256 threads
    const int blocks = (E + wavesPerBlock - 1) / wavesPerBlock;
    tp_scatter_kernel<<<blocks, 32 * wavesPerBlock, 0, stream>>>(
        x, sh, src, dst, cg, out, E);
}